// AELoss_21294447854080
// MI455X (gfx1250) — compile-verified
//
#include <hip/hip_runtime.h>
#include <math.h>

// AE (associative-embedding) loss, reference shapes:
//   N=32 batches, M=30 persons, K=17 joints, KHW = 17*256*256
// tags:   [N, KHW, 1] float32   (d_in[0])
// joints: [N, M, K, 2] int32    (d_in[1])
// out:    pushes[N] ++ pulls[N] (64 floats)
//
// Strategy: one wave32 per batch item. Lane m owns person m. The pairwise
// push term's outer product mu*mu^T is computed with V_WMMA_F32_16X16X4_F32
// (4 tiles covering a 32x32 padded person grid).

#define AE_N   32
#define AE_M   30
#define AE_K   17
#define AE_KHW (17 * 256 * 256)

typedef __attribute__((ext_vector_type(2))) float v2f;
typedef __attribute__((ext_vector_type(8))) float v8f;

__global__ __launch_bounds__(32) void ae_loss_kernel(
    const float* __restrict__ tags,
    const int*   __restrict__ joints,
    float*       __restrict__ out)
{
    const int n    = blockIdx.x;     // batch item
    const int lane = threadIdx.x;    // 0..31 (wave32, one wave per block)

    const float* tagn = tags + (size_t)n * AE_KHW;
    const int*   jn   = joints + (size_t)n * AE_M * AE_K * 2;

    // ---- per-person gather + mean (lane m == person m; lanes >= M clamped) --
    const int m = (lane < AE_M) ? lane : (AE_M - 1);

    float tv[AE_K];
    float vv[AE_K];
    float tsum = 0.0f, cnt = 0.0f;
#pragma unroll
    for (int k = 0; k < AE_K; ++k) {
        const int base = (m * AE_K + k) * 2;
        const int idx  = jn[base + 0];
        const int vis  = jn[base + 1];
        const float v  = (vis > 0) ? 1.0f : 0.0f;
        const float t  = tagn[idx];
        tv[k] = t;
        vv[k] = v;
        tsum += t * v;
        cnt  += v;
    }

    const bool  valid = (lane < AE_M) && (cnt > 0.0f);
    const float cnt_s = fmaxf(cnt, 1.0f);
    const float mu    = tsum / cnt_s;

    float pull_pp = 0.0f;
#pragma unroll
    for (int k = 0; k < AE_K; ++k) {
        const float d = tv[k] - mu;
        pull_pp += vv[k] * d * d;
    }
    pull_pp /= cnt_s;

    const float pvf  = valid ? 1.0f : 0.0f;   // person-valid mask
    const float muM  = valid ? mu : 0.0f;     // masked mu (pads WMMA operands)
    const float pull = valid ? pull_pp : 0.0f;

    // ---- publish per-person values for the pairwise stage ------------------
    __shared__ float s_mu[32];
    __shared__ float s_musq[32];
    __shared__ float s_pv[32];
    s_mu[lane]   = muM;
    s_musq[lane] = muM * muM;
    s_pv[lane]   = pvf;
    __syncthreads();

    // ---- wave reductions: n_tags and pull sum ------------------------------
    float nsum = pvf;
    float psum = pull;
#pragma unroll
    for (int off = 16; off > 0; off >>= 1) {
        nsum += __shfl_xor(nsum, off, 32);
        psum += __shfl_xor(psum, off, 32);
    }

    // ---- push term: mu outer product via V_WMMA_F32_16X16X4_F32 ------------
    // (mu_i - mu_j)^2 = musq_i + musq_j - 2 * (mu_i * mu_j)
    // 32x32 padded person grid -> 4 tiles of 16x16.
    const int nIdx    = lane & 15;          // D-tile column for this lane
    const int rowBase = (lane >> 4) << 3;   // D-tile rows: 0..7 or 8..15

    float pushAcc = 0.0f;
#pragma unroll
    for (int I = 0; I < 2; ++I) {
        // A (16x4): lanes 0-15 v0 hold K=0 column = mu[16I + lane]; rest 0.
        const float aK0 = s_mu[16 * I + nIdx];
        v2f a;
        a.x = (lane < 16) ? aK0 : 0.0f;
        a.y = 0.0f;
#pragma unroll
        for (int J = 0; J < 2; ++J) {
            // B (4x16): v0 lanes 0-15 hold row K=0 = mu[16J + lane]; rest 0.
            const float bK0 = s_mu[16 * J + nIdx];
            v2f b;
            b.x = (lane < 16) ? bK0 : 0.0f;
            b.y = 0.0f;

            v8f c = {};
            c = __builtin_amdgcn_wmma_f32_16x16x4_f32(
                    /*neg_a=*/false, a, /*neg_b=*/false, b,
                    /*c_mod=*/(short)0, c,
                    /*reuse_a=*/false, /*reuse_b=*/false);

            const float musqN = s_musq[16 * J + nIdx];
            const float pvN   = s_pv[16 * J + nIdx];
#pragma unroll
            for (int r = 0; r < 8; ++r) {
                const int   Mi    = 16 * I + rowBase + r;
                const float d2    = s_musq[Mi] + musqN - 2.0f * c[r];
                const float w     = s_pv[Mi] * pvN;
                pushAcc += w * __expf(-d2);
            }
        }
    }

#pragma unroll
    for (int off = 16; off > 0; off >>= 1) {
        pushAcc += __shfl_xor(pushAcc, off, 32);
    }

    // ---- finalize ----------------------------------------------------------
    if (lane == 0) {
        const float ntags = nsum;
        const float push  = pushAcc - ntags;  // remove diagonal (as reference)
        const float denom = fmaxf((ntags - 1.0f) * ntags, 1.0f);
        out[n]        = push / denom * 0.5f;          // pushes[n]
        out[AE_N + n] = psum / fmaxf(ntags, 1.0f);    // pulls[n]
    }
}

extern "C" void kernel_launch(void* const* d_in, const int* in_sizes, int n_in,
                              void* d_out, int out_size, void* d_ws, size_t ws_size,
                              hipStream_t stream) {
    (void)in_sizes; (void)n_in; (void)d_ws; (void)ws_size; (void)out_size;
    const float* tags   = (const float*)d_in[0];
    const int*   joints = (const int*)d_in[1];
    float*       out    = (float*)d_out;

    ae_loss_kernel<<<dim3(AE_N), dim3(32), 0, stream>>>(tags, joints, out);
}